// EntropyRegularizer_53386443489554
// MI455X (gfx1250) — compile-verified
//
#include <hip/hip_runtime.h>
#include <math.h>
#include <stdint.h>

#define EPS_F 1e-8f
#define HFEAT 1024
#define KCENT 10
#define CHUNK 256                 // features staged per iteration
#define NCHUNK (HFEAT / CHUNK)
#define ROWS 16                   // x rows per block
#define LSTR 260                  // LDS row stride (floats): 16B-aligned, conflict-free b64 reads
#define TILE_F (ROWS * LSTR)      // floats per LDS tile buffer
#define TILE_B (TILE_F * 4)       // bytes per LDS tile buffer

typedef float v2f __attribute__((ext_vector_type(2)));
typedef float v8f __attribute__((ext_vector_type(8)));
typedef unsigned int u32x4 __attribute__((ext_vector_type(4)));
typedef int i32x4 __attribute__((ext_vector_type(4)));
typedef int i32x8 __attribute__((ext_vector_type(8)));

#if __has_builtin(__builtin_amdgcn_tensor_load_to_lds) && \
    __has_builtin(__builtin_amdgcn_s_wait_tensorcnt)
#define USE_TDM 1
#else
#define USE_TDM 0
#endif

#if USE_TDM
// 2-D tile load Global -> LDS via Tensor Data Mover.
// Elements are 4-byte; LDS padding: 4 DWORDs after every 256 DWORDs
// (pad_interval code 7 = 256 DW, pad_amount code 3 = 4 DW) -> row stride 260 f32.
__device__ __forceinline__ void tdm_load_2d(unsigned lds_addr, const void* gaddr,
                                            unsigned tile_d0, unsigned tile_d1,
                                            unsigned tensor_d0, unsigned tensor_d1,
                                            unsigned row_stride_elems)
{
    unsigned long long ga = (unsigned long long)(uintptr_t)gaddr;
    u32x4 g0;
    g0.x = 1u;                                                   // count=1, user descriptor
    g0.y = lds_addr;                                             // LDS byte address
    g0.z = (unsigned)ga;                                         // global_addr[31:0]
    g0.w = (unsigned)((ga >> 32) & 0x01ffffffull) | (2u << 30);  // [56:32] | type=2
    i32x8 g1;
    g1[0] = (int)((2u << 16)      // data_size = 4 bytes
                | (1u << 20)      // pad_enable
                | (7u << 22)      // pad_interval: every 256 DWORDs
                | (3u << 25));    // pad_amount: 4 DWORDs
    g1[1] = (int)((tensor_d0 & 0xffffu) << 16);                       // td0[15:0]
    g1[2] = (int)((tensor_d0 >> 16) | ((tensor_d1 & 0xffffu) << 16)); // td0[31:16], td1[15:0]
    g1[3] = (int)((tensor_d1 >> 16) | (tile_d0 << 16));               // td1[31:16], tile_dim0
    g1[4] = (int)(tile_d1 & 0xffffu);                                 // tile_dim1 ; tile_dim2=0
    g1[5] = (int)row_stride_elems;                                    // tensor_dim0_stride[31:0]
    g1[6] = 0;
    g1[7] = 0;
    i32x4 z4 = {0, 0, 0, 0};                                          // 2-D: groups 2/3 unused
    i32x8 z8 = {0, 0, 0, 0, 0, 0, 0, 0};
    // amdgpu-toolchain (clang-23) 6-arg form
    __builtin_amdgcn_tensor_load_to_lds(g0, g1, z4, z4, z8, 0);
}
#endif

__global__ __launch_bounds__(256) void knife_zero(float* accum) {
    accum[0] = 0.0f;
}

__global__ __launch_bounds__(256) void knife_main(
    const float* __restrict__ x,        // [N, H]
    const float* __restrict__ centers,  // [K, H]
    const float* __restrict__ weights,  // [K]
    const float* __restrict__ scales,   // [K]
    float* __restrict__ accum,          // [1] global sum of -log(density+eps)
    int nrows)
{
    __shared__ float tile[2 * TILE_F];   // x rows, double buffered
    __shared__ float ctile[2 * TILE_F];  // centers (rows 10..15 zero), double buffered
    __shared__ float dtile[ROWS * 16];   // 16x16 dot accumulator
    __shared__ float xsq[ROWS];
    __shared__ float csq[16];

    const int tid  = threadIdx.x;
    const int lane = tid & 31;
    const int wave = tid >> 5;
    const long rowBase = (long)blockIdx.x * ROWS;

    // one-time zero init of shared accumulators and padded center rows (both buffers)
    for (int i = tid; i < ROWS * 16; i += 256) dtile[i] = 0.0f;
    if (tid < ROWS) xsq[tid] = 0.0f;
    if (tid < 16)   csq[tid] = 0.0f;
    for (int i = tid; i < (ROWS - KCENT) * LSTR; i += 256) {
        ctile[KCENT * LSTR + i] = 0.0f;
        ctile[TILE_F + KCENT * LSTR + i] = 0.0f;
    }

#if USE_TDM
    const unsigned tileAddr  = (unsigned)(uintptr_t)&tile[0];
    const unsigned ctileAddr = (unsigned)(uintptr_t)&ctile[0];
    // prologue: wave 0 kicks off the DMA for chunk 0 into buffer 0
    if (wave == 0) {
        tdm_load_2d(tileAddr,  x + rowBase * HFEAT, CHUNK, ROWS,
                    HFEAT, (unsigned)nrows, HFEAT);
        tdm_load_2d(ctileAddr, centers,             CHUNK, KCENT,
                    HFEAT, KCENT, HFEAT);
    }
#endif

    v8f acc = {};                 // 16x16 f32 C/D tile (8 VGPRs)
    float asq = 0.0f, bsq = 0.0f; // per-lane partial squared norms

    const int arow = lane & 15;         // row (A) / center (B) this lane holds
    const int koff = (lane >> 4) * 2;   // k sub-offset inside each 4-wide step

    for (int ch = 0; ch < NCHUNK; ++ch) {
        const int buf = ch & 1;

#if USE_TDM
        // issue next chunk's DMA into the other buffer (safe: end-of-iter barrier
        // guarantees everyone finished computing on it), then wait for this chunk.
        if (ch + 1 < NCHUNK) {
            if (wave == 0) {
                tdm_load_2d(tileAddr + (unsigned)((buf ^ 1) * TILE_B),
                            x + rowBase * HFEAT + (ch + 1) * CHUNK,
                            CHUNK, ROWS, HFEAT, (unsigned)nrows, HFEAT);
                tdm_load_2d(ctileAddr + (unsigned)((buf ^ 1) * TILE_B),
                            centers + (ch + 1) * CHUNK,
                            CHUNK, KCENT, HFEAT, KCENT, HFEAT);
                __builtin_amdgcn_s_wait_tensorcnt(2);  // only newest 2 outstanding
            }
        } else if (wave == 0) {
            __builtin_amdgcn_s_wait_tensorcnt(0);
        }
#else
        __syncthreads();  // previous chunk's compute done before overwrite
        // stage x tile: 16 rows x 256 cols, coalesced float4
#pragma unroll
        for (int it = 0; it < 4; ++it) {
            int f = it * 256 + tid;        // float4 index; 64 per row
            int r = f >> 6, c4 = f & 63;
            float4 v = *(const float4*)(x + (rowBase + r) * HFEAT + ch * CHUNK + c4 * 4);
            *(float4*)(tile + buf * TILE_F + r * LSTR + c4 * 4) = v;
        }
        // stage centers slice: 10 rows x 256 cols
        for (int f = tid; f < KCENT * 64; f += 256) {
            int r = f >> 6, c4 = f & 63;
            float4 v = *(const float4*)(centers + (long)r * HFEAT + ch * CHUNK + c4 * 4);
            *(float4*)(ctile + buf * TILE_F + r * LSTR + c4 * 4) = v;
        }
#endif
        __syncthreads();  // chunk data visible to all waves

        const float* tb = tile + buf * TILE_F;
        const float* cb = ctile + buf * TILE_F;
        // this wave's 32-feature slice of the chunk: 8 WMMA k-steps of 4
#pragma unroll
        for (int t = 0; t < 8; ++t) {
            int col = wave * 32 + t * 4 + koff;
            v2f a = *(const v2f*)(tb + arow * LSTR + col);
            v2f b = *(const v2f*)(cb + arow * LSTR + col);
            asq += a.x * a.x + a.y * a.y;
            bsq += b.x * b.x + b.y * b.y;
            // D = A(16x4 f32) x B(4x16 f32) + C
            acc = __builtin_amdgcn_wmma_f32_16x16x4_f32(
                false, a, false, b, (short)0, acc, false, false);
        }

#if USE_TDM
        __syncthreads();  // all waves done with buf before TDM refills it
#endif
    }

    // fold per-lane partial norms and per-wave partial dot tiles into LDS
    atomicAdd(&xsq[arow], asq);
    atomicAdd(&csq[arow], bsq);
#pragma unroll
    for (int i = 0; i < 8; ++i) {
        int m = i + ((lane >> 4) << 3);     // D layout: VGPR i holds rows i / i+8
        atomicAdd(&dtile[m * 16 + arow], acc[i]);
    }
    __syncthreads();

    // epilogue: wave 0 computes densities for the 16 rows
    if (wave == 0) {
        int j = arow;                        // center index for this lane
        float cs = csq[j];
        float w = 0.0f, inv2s2 = 0.0f;
        if (j < KCENT) {
            w = weights[j];
            float s = scales[j];
            inv2s2 = 1.0f / (2.0f * s * s);
        }
        float sumlog = 0.0f;
#pragma unroll
        for (int i = 0; i < 8; ++i) {
            int m = i + ((lane >> 4) << 3);
            float dot = dtile[m * 16 + j];
            float d2 = xsq[m] + cs - 2.0f * dot;
            d2 = fmaxf(d2, 0.0f);
            float val = w * __expf(-d2 * inv2s2);   // zero for padded centers
            // reduce over the 16 centers (within each 16-lane half)
            val += __shfl_xor(val, 1, 16);
            val += __shfl_xor(val, 2, 16);
            val += __shfl_xor(val, 4, 16);
            val += __shfl_xor(val, 8, 16);
            sumlog += -__logf(val + EPS_F);
        }
        sumlog += __shfl_xor(sumlog, 16, 32);        // rows 0-7 + rows 8-15
        if (lane == 0) atomicAdd(accum, sumlog);
    }
}

__global__ void knife_final(const float* __restrict__ accum,
                            float* __restrict__ out, float invN) {
    float h = accum[0] * invN;       // h_entropy
    float e = h;                     // BETA * h, BETA = 1
    float t = h * h;                 // (h - TARGET)^2, TARGET = 0
    out[0] = e;
    out[1] = t;
    out[2] = e + t;
    out[3] = h;
}

extern "C" void kernel_launch(void* const* d_in, const int* in_sizes, int n_in,
                              void* d_out, int out_size, void* d_ws, size_t ws_size,
                              hipStream_t stream) {
    const float* x       = (const float*)d_in[0];
    const float* centers = (const float*)d_in[1];
    const float* weights = (const float*)d_in[2];
    const float* scales  = (const float*)d_in[3];
    float* accum = (float*)d_ws;
    float* out   = (float*)d_out;

    const int N = in_sizes[0] / HFEAT;   // 8192 rows

    knife_zero<<<1, 1, 0, stream>>>(accum);
    knife_main<<<N / ROWS, 256, 0, stream>>>(x, centers, weights, scales, accum, N);
    knife_final<<<1, 1, 0, stream>>>(accum, out, 1.0f / (float)N);
}